// ChebConv_26216480375298
// MI455X (gfx1250) — compile-verified
//
#include <hip/hip_runtime.h>
#include <hip/hip_bf16.h>

typedef __attribute__((ext_vector_type(2))) float v2f;
typedef __attribute__((ext_vector_type(8))) float v8f;

#define F_DIM 128LL
#define APITCH 132   // LDS row pitch (floats): 4*m+k bank pattern -> conflict-free

// ---------- 1. degree accumulation ----------
__global__ void degree_kernel(const int* __restrict__ dst, float* __restrict__ deg,
                              long long E) {
    long long e = (long long)blockIdx.x * blockDim.x + threadIdx.x;
    if (e < E) atomicAdd(&deg[dst[e]], 1.0f);
}

// ---------- 2. norm = deg^{-1/2} (in place) ----------
__global__ void norm_kernel(float* __restrict__ deg, long long N) {
    long long i = (long long)blockIdx.x * blockDim.x + threadIdx.x;
    if (i < N) deg[i] = rsqrtf(deg[i]);
}

// ---------- 3. SpMM: agg[dst] += h[src] * norm[src]  (1 edge per 32-lane wave) ----------
__global__ void spmm_scatter(const float* __restrict__ h, const float* __restrict__ norm,
                             const int* __restrict__ src, const int* __restrict__ dst,
                             float* __restrict__ agg, long long E) {
    long long t = (long long)blockIdx.x * blockDim.x + threadIdx.x;
    long long e = t >> 5;
    if (e >= E) return;
    int lane = (int)(t & 31);
    int s = src[e];
    int d = dst[e];
    float ns = norm[s];
    const float4* hv = (const float4*)(h + (long long)s * F_DIM);
    float4 v = hv[lane];                       // 32 lanes x float4 = 128 floats
    float* outp = agg + (long long)d * F_DIM + lane * 4;
    atomicAdd(outp + 0, v.x * ns);
    atomicAdd(outp + 1, v.y * ns);
    atomicAdd(outp + 2, v.z * ns);
    atomicAdd(outp + 3, v.w * ns);
}

// ---------- 4. Tx1 = agg*norm*lap - Tx0   (in place over agg) ----------
__global__ void cheb1_kernel(float* __restrict__ buf1, const float* __restrict__ feat,
                             const float* __restrict__ norm,
                             const float* __restrict__ lambda_max, long long NF) {
    long long i = (long long)blockIdx.x * blockDim.x + threadIdx.x;
    if (i >= NF) return;
    float lap = 2.0f / lambda_max[0];
    long long row = i >> 7;                    // /128
    buf1[i] = buf1[i] * norm[row] * lap - feat[i];
}

// ---------- 5. Tx2 = 2*(agg*norm*lap - Tx1) - Tx0  (in place over agg) ----------
__global__ void cheb2_kernel(float* __restrict__ buf2, const float* __restrict__ buf1,
                             const float* __restrict__ feat, const float* __restrict__ norm,
                             const float* __restrict__ lambda_max, long long NF) {
    long long i = (long long)blockIdx.x * blockDim.x + threadIdx.x;
    if (i >= NF) return;
    float lap = 2.0f / lambda_max[0];
    long long row = i >> 7;
    buf2[i] = 2.0f * (buf2[i] * norm[row] * lap - buf1[i]) - feat[i];
}

// ---------- 6. out = Tx0@W0 + Tx1@W1 + Tx2@W2 + b   (WMMA f32 16x16x4) ----------
// Block = 256 threads = 8 waves. The block owns one 16-row tile; each wave owns
// one 16-col tile of F_OUT=128. The three 16x128 A tiles are staged once in LDS
// (shared by all 8 waves -> kills 8x redundant global A traffic), then WMMA
// fragments come from conflict-free ds_load_b64.
__global__ __launch_bounds__(256) void cheb_gemm_wmma(
    const float* __restrict__ tx0, const float* __restrict__ tx1,
    const float* __restrict__ tx2, const float* __restrict__ W,
    const float* __restrict__ bias, float* __restrict__ out) {
    __shared__ float sA[3][16][APITCH];

    const int tid   = threadIdx.x;
    const int lane  = tid & 31;
    const int wave  = tid >> 5;                // col tile 0..7
    const int row0  = blockIdx.x * 16;
    const int col0  = wave * 16;
    const int mn    = lane & 15;               // M index for A, N index for B/C
    const int khalf = lane >> 4;               // 0 or 1

    // Cooperative stage of A tiles: 3 x 16 x 128 floats = 512 float4 per matrix,
    // 2 float4 per thread per matrix. 16B-aligned LDS stores (pitch 528B).
    const float* mats[3] = {tx0, tx1, tx2};
#pragma unroll
    for (int m = 0; m < 3; ++m) {
#pragma unroll
        for (int i = tid; i < 512; i += 256) {
            int r  = i >> 5;                   // 32 float4 per row
            int c4 = i & 31;
            float4 v = ((const float4*)(mats[m] + (long long)(row0 + r) * F_DIM))[c4];
            float* dp = &sA[m][r][c4 * 4];
            dp[0] = v.x; dp[1] = v.y; dp[2] = v.z; dp[3] = v.w;
        }
    }
    __syncthreads();

    v8f acc = {};
#pragma unroll
    for (int m = 0; m < 3; ++m) {
        const float* Bm = W + (long long)m * F_DIM * F_DIM + col0 + mn;
#pragma unroll 4
        for (int kk = 0; kk < 128; kk += 4) {
            int kb = kk + 2 * khalf;
            // A fragment: lane holds (M=mn, K=kb..kb+1) -> ds_load_b64, banks 4*mn+kb+{0,1}
            v2f a;
            a.x = sA[m][mn][kb];
            a.y = sA[m][mn][kb + 1];
            // B fragment: lane holds (K=kb..kb+1, N=col0+mn) from L2-resident W
            v2f b;
            b.x = Bm[(long long)kb * F_DIM];
            b.y = Bm[(long long)(kb + 1) * F_DIM];
            acc = __builtin_amdgcn_wmma_f32_16x16x4_f32(
                false, a, false, b, (short)0, acc, false, false);
        }
    }

    // C layout: VGPR v holds (M = v + 8*khalf, N = mn)
    float bv = bias[col0 + mn];
#pragma unroll
    for (int v = 0; v < 8; ++v) {
        int M = v + 8 * khalf;
        out[(long long)(row0 + M) * F_DIM + col0 + mn] = acc[v] + bv;
    }
}

extern "C" void kernel_launch(void* const* d_in, const int* in_sizes, int n_in,
                              void* d_out, int out_size, void* d_ws, size_t ws_size,
                              hipStream_t stream) {
    const float* feat       = (const float*)d_in[0];
    const int*   src        = (const int*)d_in[1];
    const int*   dst        = (const int*)d_in[2];
    const float* W          = (const float*)d_in[3];
    const float* bias       = (const float*)d_in[4];
    const float* lambda_max = (const float*)d_in[5];

    const long long N  = (long long)in_sizes[0] / F_DIM;   // 100000
    const long long E  = (long long)in_sizes[1];           // 3200000
    const long long NF = N * F_DIM;

    // workspace layout: norm[N] (256-aligned) | buf1[N*128] | buf2[N*128]
    float* norm = (float*)d_ws;
    float* buf1 = norm + ((N + 255LL) & ~255LL);
    float* buf2 = buf1 + NF;

    const int T = 256;

    // degree -> norm
    hipMemsetAsync(norm, 0, N * sizeof(float), stream);
    degree_kernel<<<(unsigned)((E + T - 1) / T), T, 0, stream>>>(dst, norm, E);
    norm_kernel<<<(unsigned)((N + T - 1) / T), T, 0, stream>>>(norm, N);

    // Tx1
    hipMemsetAsync(buf1, 0, NF * sizeof(float), stream);
    spmm_scatter<<<(unsigned)((E * 32 + T - 1) / T), T, 0, stream>>>(feat, norm, src, dst, buf1, E);
    cheb1_kernel<<<(unsigned)((NF + T - 1) / T), T, 0, stream>>>(buf1, feat, norm, lambda_max, NF);

    // Tx2
    hipMemsetAsync(buf2, 0, NF * sizeof(float), stream);
    spmm_scatter<<<(unsigned)((E * 32 + T - 1) / T), T, 0, stream>>>(buf1, norm, src, dst, buf2, E);
    cheb2_kernel<<<(unsigned)((NF + T - 1) / T), T, 0, stream>>>(buf2, buf1, feat, norm, lambda_max, NF);

    // projection GEMM with WMMA (N multiple of 16: 100000/16 = 6250)
    cheb_gemm_wmma<<<(unsigned)(N / 16), 256, 0, stream>>>(feat, buf1, buf2, W, bias, (float*)d_out);
}